// ModifiedEGNN_Network_33921651703918
// MI455X (gfx1250) — compile-verified
//
#include <hip/hip_runtime.h>
#include <hip/hip_bf16.h>

// ---------------------------------------------------------------------------
// ModifiedEGNN: y = x@W_lin^T + b ; mean-scatter over edges (+self loops) ;
// h = relu(agg@W1^T + b1) ; out = sum_n (h@W2^T + b2)  ==  (sum_n h)@W2^T + N*b2
// ---------------------------------------------------------------------------

typedef float v2f __attribute__((ext_vector_type(2)));
typedef float v8f __attribute__((ext_vector_type(8)));

static constexpr int DK = 128;   // feature dim D
static constexpr int DH = 128;   // hidden dim

// ---------------------------------------------------------------------------
// GEMM: Y[m,n] = sum_k X[m,k] * W[n,k] + bias[n]
// One block = 16 output rows; 8 waves, wave w owns columns [16w, 16w+16).
// Uses V_WMMA_F32_16X16X4_F32 (full f32 — the GEMMs are tiny vs the scatter,
// so there is no benefit to dropping precision).
// A layout (16x4 f32): lane L -> row M=L&15, VGPR0={K=0|K=2}, VGPR1={K=1|K=3}
// C/D layout: VGPR v -> row v + 8*(lane>=16), col = lane&15
// ---------------------------------------------------------------------------
__global__ void egnn_gemm_lin(const float* __restrict__ X,
                              const float* __restrict__ W,
                              const float* __restrict__ bias,
                              float* __restrict__ Y) {
  const int lane = threadIdx.x & 31;
  const int wave = threadIdx.x >> 5;
  const int m0   = blockIdx.x << 4;
  const int n0   = wave << 4;
  const int mrow = m0 + (lane & 15);
  const int ncol = n0 + (lane & 15);
  const int kb   = (lane >> 4) << 1;     // 0 or 2

  const float* arow = X + (size_t)mrow * DK + kb;
  const float* brow = W + (size_t)ncol * DK + kb;

  v8f acc = {};
#pragma unroll 8
  for (int k0 = 0; k0 < DK; k0 += 4) {
    v2f a = *(const v2f*)(arow + k0);
    v2f b = *(const v2f*)(brow + k0);
    acc = __builtin_amdgcn_wmma_f32_16x16x4_f32(
        /*neg_a=*/false, a, /*neg_b=*/false, b,
        /*c_mod=*/(short)0, acc, /*reuse_a=*/false, /*reuse_b=*/false);
  }

  const int   rbase = m0 + ((lane >> 4) << 3);
  const float bv    = bias[ncol];
#pragma unroll
  for (int v = 0; v < 8; ++v) {
    Y[(size_t)(rbase + v) * DK + ncol] = acc[v] + bv;
  }
}

// ---------------------------------------------------------------------------
// Edge scatter: one wave per (edge or self-loop); lane handles 4 features.
// summ[dst] += y[src]; cnt[dst] += 1. Edges [0,E) come from edge_index,
// elements [E, E+N) are the self loops (src=dst=e-E).
// ---------------------------------------------------------------------------
__global__ void egnn_scatter(const int* __restrict__ esrc,
                             const int* __restrict__ edst,
                             const float* __restrict__ Y,
                             float* __restrict__ summ,
                             float* __restrict__ cnt,
                             int E, int Nnodes) {
  const int lane = threadIdx.x & 31;
  const int e    = (int)((blockIdx.x * blockDim.x + threadIdx.x) >> 5);
  if (e >= E + Nnodes) return;             // wave-uniform exit

  int s, d;
  if (e < E) { s = esrc[e]; d = edst[e]; }
  else       { s = d = e - E; }

  const float4 v = *(const float4*)(Y + (size_t)s * DK + (lane << 2));
  float* o = summ + (size_t)d * DK + (lane << 2);
  atomicAdd(o + 0, v.x);
  atomicAdd(o + 1, v.y);
  atomicAdd(o + 2, v.z);
  atomicAdd(o + 3, v.w);
  if (lane == 0) atomicAdd(&cnt[d], 1.0f);
}

// ---------------------------------------------------------------------------
// Fused: agg = summ / max(cnt,1); h = relu(agg@W1^T + b1); hsum += sum_rows(h)
// Same WMMA tiling as egnn_gemm_lin. Row scale folds into the A operand
// (each lane only ever supplies elements of row M = lane&15).
// Column partials are folded across the two half-waves with a lane shuffle,
// then 16 atomics per wave accumulate into hsum[128].
// ---------------------------------------------------------------------------
__global__ void egnn_fc1_reduce(const float* __restrict__ summ,
                                const float* __restrict__ cnt,
                                const float* __restrict__ W1,
                                const float* __restrict__ b1,
                                float* __restrict__ hsum) {
  const int lane = threadIdx.x & 31;
  const int wave = threadIdx.x >> 5;
  const int m0   = blockIdx.x << 4;
  const int n0   = wave << 4;
  const int mrow = m0 + (lane & 15);
  const int ncol = n0 + (lane & 15);
  const int kb   = (lane >> 4) << 1;

  const float inv = 1.0f / fmaxf(cnt[mrow], 1.0f);
  const float* arow = summ + (size_t)mrow * DK + kb;
  const float* brow = W1 + (size_t)ncol * DK + kb;

  v8f acc = {};
#pragma unroll 8
  for (int k0 = 0; k0 < DK; k0 += 4) {
    v2f a = *(const v2f*)(arow + k0);
    a.x *= inv;
    a.y *= inv;
    v2f b = *(const v2f*)(brow + k0);
    acc = __builtin_amdgcn_wmma_f32_16x16x4_f32(
        false, a, false, b, (short)0, acc, false, false);
  }

  const float bv = b1[ncol];
  float part = 0.0f;
#pragma unroll
  for (int v = 0; v < 8; ++v) part += fmaxf(acc[v] + bv, 0.0f);

  // lane L and L+16 hold the same column (rows m0..m0+7 vs m0+8..m0+15)
  part += __shfl_xor(part, 16, 32);
  if (lane < 16) atomicAdd(&hsum[ncol], part);
}

// ---------------------------------------------------------------------------
// Final: out[o] = hsum . W2[o,:] + N * b2[o], o in [0,3)
// ---------------------------------------------------------------------------
__global__ void egnn_final(const float* __restrict__ hsum,
                           const float* __restrict__ W2,
                           const float* __restrict__ b2,
                           float* __restrict__ out,
                           int Nnodes) {
  const int o = threadIdx.x;
  if (o < 3) {
    float s = b2[o] * (float)Nnodes;
#pragma unroll 8
    for (int d = 0; d < DH; ++d) s += hsum[d] * W2[o * DH + d];
    out[o] = s;
  }
}

// ---------------------------------------------------------------------------
extern "C" void kernel_launch(void* const* d_in, const int* in_sizes, int n_in,
                              void* d_out, int out_size, void* d_ws, size_t ws_size,
                              hipStream_t stream) {
  const float* x     = (const float*)d_in[0];
  const int*   eidx  = (const int*)d_in[1];    // [2, E] flat
  const float* W_lin = (const float*)d_in[2];
  const float* b_lin = (const float*)d_in[3];
  const float* W1    = (const float*)d_in[4];
  const float* b1    = (const float*)d_in[5];
  const float* W2    = (const float*)d_in[6];
  const float* b2    = (const float*)d_in[7];

  const int N = in_sizes[0] / DK;   // 50000 (divisible by 16: 3125 tiles)
  const int E = in_sizes[1] / 2;    // 600000

  // Workspace layout (floats): y[N*DK] | summ[N*DK] | cnt[N] | hsum[DH]
  float* ws   = (float*)d_ws;
  float* y    = ws;
  float* summ = ws + (size_t)N * DK;
  float* cnt  = summ + (size_t)N * DK;
  float* hs   = cnt + N;

  // Zero the accumulation buffers (summ, cnt, hsum are contiguous).
  hipMemsetAsync(summ, 0, ((size_t)N * DK + N + DH) * sizeof(float), stream);

  // 1) y = x @ W_lin^T + b_lin
  egnn_gemm_lin<<<dim3(N / 16), dim3(256), 0, stream>>>(x, W_lin, b_lin, y);

  // 2) scatter-mean numerator/denominator over edges + self loops
  const int total  = E + N;
  const int blocks = (total + 7) / 8;        // 8 waves per 256-thread block
  egnn_scatter<<<dim3(blocks), dim3(256), 0, stream>>>(
      eidx, eidx + E, y, summ, cnt, E, N);

  // 3) mean + fc1 + relu + column reduction into hsum[128]
  egnn_fc1_reduce<<<dim3(N / 16), dim3(256), 0, stream>>>(summ, cnt, W1, b1, hs);

  // 4) out = hsum @ W2^T + N*b2
  egnn_final<<<dim3(1), dim3(32), 0, stream>>>(hs, W2, b2, (float*)d_out, N);
}